// EfficientMemorySoftmax_82248623718484
// MI455X (gfx1250) — compile-verified
//
#include <hip/hip_runtime.h>
#include <cstdint>

// Softmax over last axis of [4,16,1024,1024] fp32 -> 65536 rows x 1024 cols.
// Bandwidth-bound: 512MB traffic / 23.3 TB/s ~= 22us floor.
// One wave32 per row; row staged to LDS via CDNA5 async global->LDS DMA,
// then fully register-resident single-pass softmax with NT stores.

#define ROW_LEN 1024
#define ROW_F4  (ROW_LEN / 4)          // 256 float4 per row
#define WAVES_PER_BLOCK 8
#define BLOCK_THREADS (WAVES_PER_BLOCK * 32)

// Native clang vector type: accepted by __builtin_nontemporal_* and lowers to b128.
typedef __attribute__((ext_vector_type(4))) float v4f;

__global__ __launch_bounds__(BLOCK_THREADS)
void softmax_rows_async(const float* __restrict__ x,
                        float* __restrict__ y,
                        int nrows) {
  // 8 waves * 4KB row = 32KB LDS per block (WGP has 320KB)
  __shared__ v4f sbuf[WAVES_PER_BLOCK * ROW_F4];

  const int lane = threadIdx.x & 31;
  const int wave = threadIdx.x >> 5;
  const int row  = blockIdx.x * WAVES_PER_BLOCK + wave;
  if (row >= nrows) return;  // wave-uniform

  const v4f* wbuf = &sbuf[wave * ROW_F4];
  // Low 32 bits of a flat shared-aperture pointer == LDS byte address.
  const uint32_t lds_base = (uint32_t)(uintptr_t)wbuf;
  // Input is 256MB -> byte offsets fit in u32 (GVS mode: saddr base + u32 vgpr offset).
  const uint32_t gbyte_base = (uint32_t)row * (uint32_t)(ROW_LEN * sizeof(float));

  // ---- Stage row into LDS with async global->LDS b128 copies (ASYNCcnt) ----
  // Lane-strided: instruction k copies bytes [k*512, k*512+512) of the row,
  // 16B per lane, contiguous 512B per wave-instruction.
#pragma unroll
  for (int k = 0; k < 8; ++k) {
    const uint32_t idx16    = (uint32_t)(k * 32 + lane) * 16u;
    const uint32_t lds_addr = lds_base + idx16;
    const uint32_t goff     = gbyte_base + idx16;
    asm volatile("global_load_async_to_lds_b128 %0, %1, %2 th:TH_LOAD_NT"
                 :: "v"(lds_addr), "v"(goff), "s"(x)
                 : "memory");
  }
  asm volatile("s_wait_asynccnt 0" ::: "memory");

  // ---- Pull 32 values/lane from LDS (ds_load_b128, conflict-free) ----
  v4f v[8];
#pragma unroll
  for (int k = 0; k < 8; ++k) {
    v[k] = wbuf[k * 32 + lane];
  }

  // ---- Row max: lane-local then 5-step xor-shuffle wave32 reduction ----
  float m = v[0].x;
#pragma unroll
  for (int k = 0; k < 8; ++k) {
    m = fmaxf(m, v[k].x);
    m = fmaxf(m, v[k].y);
    m = fmaxf(m, v[k].z);
    m = fmaxf(m, v[k].w);
  }
#pragma unroll
  for (int off = 16; off >= 1; off >>= 1) {
    m = fmaxf(m, __shfl_xor(m, off, 32));
  }

  // ---- exp(x - m) and row sum ----
  float s = 0.0f;
#pragma unroll
  for (int k = 0; k < 8; ++k) {
    v[k].x = __expf(v[k].x - m);
    v[k].y = __expf(v[k].y - m);
    v[k].z = __expf(v[k].z - m);
    v[k].w = __expf(v[k].w - m);
    s += v[k].x + v[k].y + v[k].z + v[k].w;
  }
#pragma unroll
  for (int off = 16; off >= 1; off >>= 1) {
    s += __shfl_xor(s, off, 32);
  }
  const float inv = 1.0f / s;

  // ---- Scale and stream out with non-temporal b128 stores ----
  v4f* __restrict__ y4 = (v4f*)y;
  const int out_base = row * ROW_F4;
#pragma unroll
  for (int k = 0; k < 8; ++k) {
    v4f o = v[k] * inv;
    __builtin_nontemporal_store(o, &y4[out_base + k * 32 + lane]);
  }
}

extern "C" void kernel_launch(void* const* d_in, const int* in_sizes, int n_in,
                              void* d_out, int out_size, void* d_ws, size_t ws_size,
                              hipStream_t stream) {
  const float* x = (const float*)d_in[0];
  float* y = (float*)d_out;
  const int total = in_sizes[0];
  const int nrows = total / ROW_LEN;           // 65536 for [4,16,1024,1024]
  const int blocks = (nrows + WAVES_PER_BLOCK - 1) / WAVES_PER_BLOCK;
  softmax_rows_async<<<blocks, BLOCK_THREADS, 0, stream>>>(x, y, nrows);
}